// Morph2d_18133351923968
// MI455X (gfx1250) — compile-verified
//
#include <hip/hip_runtime.h>

#define KSZ 3
#define TILE_H 32
#define TILE_W 64
#define IN_H 36           // TILE_H + 4 (5x5 total support)
#define IN_W 68           // TILE_W + 4
#define IN_STRIDE 68
#define ED_H 34           // TILE_H + 2 (ero/dil halo for pass B)
#define ED_W 66           // TILE_W + 2
#define ED_STRIDE 68      // even stride -> aligned b64 LDS loads
#define IMG_H 512
#define IMG_W 512
#define REGION 508        // computed region (H-K-1)
#define OUTW 510          // output spatial size (H-K+1)
#define NGROUP 16
#define GPG 4             // groups per workgroup
#define CHUNKS (NGROUP / GPG)
#define PLANE (OUTW * OUTW)

#if defined(__HIP_DEVICE_COMPILE__) && \
    __has_builtin(__builtin_amdgcn_global_load_async_to_lds_b128) && \
    __has_builtin(__builtin_amdgcn_s_wait_asynccnt)
#define USE_ASYNC_LDS 1
typedef int v4i __attribute__((ext_vector_type(4)));
typedef __attribute__((address_space(1))) v4i* global_v4i_ptr;
typedef __attribute__((address_space(3))) v4i* lds_v4i_ptr;
#else
#define USE_ASYNC_LDS 0
#endif

__global__ __launch_bounds__(256)
void morph2d_fused(const float* __restrict__ x,
                   const float* __restrict__ w,
                   float* __restrict__ out)
{
    __shared__ __align__(16) float sIn[IN_H * IN_STRIDE];
    __shared__ __align__(16) float sDil[ED_H * ED_STRIDE];
    __shared__ __align__(16) float sEro[ED_H * ED_STRIDE];

    const int tid = threadIdx.x;
    const int tx0 = blockIdx.x * TILE_W;
    const int ty0 = blockIdx.y * TILE_H;
    const int b   = blockIdx.z / CHUNKS;
    const int g0  = (blockIdx.z % CHUNKS) * GPG;

    const float* __restrict__ xb = x + (size_t)b * IMG_H * IMG_W;

    // ---- Stage input tile (36 x 68 floats) into LDS ----
    // All inputs that feed a *valid* (y,x < 508) position are in-bounds, so
    // clamped addresses only feed positions that pass A forces to zero anyway.
#if USE_ASYNC_LDS
    for (int i = tid; i < IN_H * (IN_W / 4); i += 256) {
        int py = i / (IN_W / 4);
        int qx = (i - py * (IN_W / 4)) * 4;
        int iy = ty0 + py; iy = iy > IMG_H - 1 ? IMG_H - 1 : iy;
        int ix = tx0 + qx; ix = ix > IMG_W - 4 ? IMG_W - 4 : ix;   // keep 16B align
        const float* gp = xb + (size_t)iy * IMG_W + ix;
        float* lp = &sIn[py * IN_STRIDE + qx];
        __builtin_amdgcn_global_load_async_to_lds_b128(
            (global_v4i_ptr)gp, (lds_v4i_ptr)lp, 0, 0);
    }
    __builtin_amdgcn_s_wait_asynccnt(0);
#else
    for (int i = tid; i < IN_H * IN_W; i += 256) {
        int py = i / IN_W;
        int px = i - py * IN_W;
        int iy = ty0 + py; iy = iy > IMG_H - 1 ? IMG_H - 1 : iy;
        int ix = tx0 + px; ix = ix > IMG_W - 1 ? IMG_W - 1 : ix;
        sIn[py * IN_STRIDE + px] = xb[(size_t)iy * IMG_W + ix];
    }
#endif
    __syncthreads();

    for (int g = g0; g < g0 + GPG; ++g) {
        // Uniform per-group weights -> scalar loads
        float wgt[KSZ][KSZ];
        #pragma unroll
        for (int i = 0; i < KSZ; ++i)
            #pragma unroll
            for (int j = 0; j < KSZ; ++j)
                wgt[i][j] = w[g * 9 + i * 3 + j];

        // ---- Pass A: dilation / erosion into LDS (34 x 66 region) ----
        for (int i = tid; i < ED_H * ED_W; i += 256) {
            int py = i / ED_W;
            int px = i - py * ED_W;
            int gy = ty0 + py, gx = tx0 + px;
            float d = 0.0f;                   // terms are |.| >= 0
            float e = __builtin_inff();
            if (gy < REGION && gx < REGION) {
                #pragma unroll
                for (int r = 0; r < KSZ; ++r) {
                    const float* p = &sIn[(py + r) * IN_STRIDE + px];
                    #pragma unroll
                    for (int c = 0; c < KSZ; ++c) {
                        float v = p[c];
                        d = fmaxf(d, fabsf(v * wgt[r][c]));
                        e = fminf(e, fabsf(v + wgt[r][c]));
                    }
                }
            } else {
                e = 0.0f;                     // zero padding of ero_full/dil_full
            }
            sDil[py * ED_STRIDE + px] = d;
            sEro[py * ED_STRIDE + px] = e;
        }
        __syncthreads();

        // ---- Pass B: opening / closing; stream 4 channels to HBM ----
        float* __restrict__ outg = out + ((size_t)b * 4 * NGROUP + 4 * g) * PLANE;
        #pragma unroll
        for (int it = 0; it < (TILE_H * TILE_W / 2) / 256; ++it) {   // 4 iters
            int task = tid + it * 256;
            int ly   = task >> 5;          // 0..31
            int lx0  = (task & 31) * 2;    // even -> aligned b64 LDS loads
            float er[KSZ][4], dl[KSZ][4];
            #pragma unroll
            for (int r = 0; r < KSZ; ++r) {
                const float2 ea = *reinterpret_cast<const float2*>(&sEro[(ly + r) * ED_STRIDE + lx0]);
                const float2 eb = *reinterpret_cast<const float2*>(&sEro[(ly + r) * ED_STRIDE + lx0 + 2]);
                const float2 da = *reinterpret_cast<const float2*>(&sDil[(ly + r) * ED_STRIDE + lx0]);
                const float2 db = *reinterpret_cast<const float2*>(&sDil[(ly + r) * ED_STRIDE + lx0 + 2]);
                er[r][0] = ea.x; er[r][1] = ea.y; er[r][2] = eb.x; er[r][3] = eb.y;
                dl[r][0] = da.x; dl[r][1] = da.y; dl[r][2] = db.x; dl[r][3] = db.y;
            }
            #pragma unroll
            for (int px = 0; px < 2; ++px) {
                int gy = ty0 + ly, gx = tx0 + lx0 + px;
                if (gy >= OUTW || gx >= OUTW) continue;
                float op = 0.0f;
                float cl = __builtin_inff();
                #pragma unroll
                for (int r = 0; r < KSZ; ++r)
                    #pragma unroll
                    for (int c = 0; c < KSZ; ++c) {
                        op = fmaxf(op, fabsf(er[r][px + c] * wgt[r][c]));
                        cl = fminf(cl, fabsf(dl[r][px + c] + wgt[r][c]));
                    }
                float dil = dl[0][px];
                float ero = er[0][px];
                if (gy >= REGION || gx >= REGION) { dil = 0.f; ero = 0.f; op = 0.f; cl = 0.f; }
                size_t o = (size_t)gy * OUTW + gx;
                __builtin_nontemporal_store(dil, outg + 0 * PLANE + o);
                __builtin_nontemporal_store(ero, outg + 1 * PLANE + o);
                __builtin_nontemporal_store(op,  outg + 2 * PLANE + o);
                __builtin_nontemporal_store(cl,  outg + 3 * PLANE + o);
            }
        }
        __syncthreads();   // WAR: next group rewrites sDil/sEro
    }
}

extern "C" void kernel_launch(void* const* d_in, const int* in_sizes, int n_in,
                              void* d_out, int out_size, void* d_ws, size_t ws_size,
                              hipStream_t stream) {
    (void)in_sizes; (void)n_in; (void)out_size; (void)d_ws; (void)ws_size;
    const float* x = (const float*)d_in[0];     // (4,1,512,512)
    const float* w = (const float*)d_in[1];     // (16,1,3,3)
    float* out = (float*)d_out;                 // (4,64,510,510)
    dim3 grid(IMG_W / TILE_W,      // 8 tiles in x  (covers 0..511)
              IMG_H / TILE_H,      // 16 tiles in y
              4 * CHUNKS);         // batch * group-chunks = 16
    morph2d_fused<<<grid, dim3(256), 0, stream>>>(x, w, out);
}